// NEQUIPLayer_26345329394135
// MI455X (gfx1250) — compile-verified
//
#include <hip/hip_runtime.h>
#include <math.h>

// ---------------------------------------------------------------------------
// NequIP layer for MI455X (gfx1250), wave32, WMMA f32 16x16x4.
// Problem is data-movement bound (~1.8GB edge gather/scatter vs ~20 GFLOP),
// so: fuse radial MLP into the edge kernel (radial weights live in 320KB LDS,
// never hit HBM), keep message accumulators L2-resident (90MB < 192MB L2),
// lane==channel mapping for coalesced gathers and f32 atomics.
// w_r1/w_r2 are staged into LDS with GLOBAL_LOAD_ASYNC_TO_LDS_B128
// (overlapped with per-edge geometry), drained with s_wait_asynccnt.
// ---------------------------------------------------------------------------

typedef float v2f __attribute__((ext_vector_type(2)));
typedef float v8f __attribute__((ext_vector_type(8)));
typedef int   v4i __attribute__((ext_vector_type(4)));

#define NN 20000      // nodes
#define NE 320000     // edges
#define CC 32         // channels (== wave32 width)
#define RD 8          // radial basis
#define HID 64
#define NP 11         // paths
#define RWN (NP*CC)   // 352
#define CUT 5.0f
#define EPS_W 0.25f
#define SILU_C 0.59601f          // sqrt(E[silu(z)^2]), z~N(0,1)
#define INV_C 0.17677670f        // 1/sqrt(32)

#if __has_builtin(__builtin_amdgcn_global_load_async_to_lds_b128)
#define HAVE_ASYNC_LDS 1
#endif

typedef __attribute__((address_space(1))) v4i gv4i;   // global int4
typedef __attribute__((address_space(3))) v4i lv4i;   // LDS int4

__device__ __constant__ float c_B[5][3][3] = {
  {{0.f,0.70710678f,0.f},{0.70710678f,0.f,0.f},{0.f,0.f,0.f}},
  {{0.f,0.f,0.f},{0.f,0.f,0.70710678f},{0.f,0.70710678f,0.f}},
  {{-0.40824829f,0.f,0.f},{0.f,-0.40824829f,0.f},{0.f,0.f,0.81649658f}},
  {{0.f,0.f,0.70710678f},{0.f,0.f,0.f},{0.70710678f,0.f,0.f}},
  {{0.70710678f,0.f,0.f},{0.f,-0.70710678f,0.f},{0.f,0.f,0.f}}
};

__device__ __forceinline__ float nsilu(float x) {
  return (x / (1.f + __expf(-x))) * (1.f / SILU_C);
}

__device__ __forceinline__ v8f wmma4(v2f a, v2f b, v8f c) {
  // D(16x16,f32) += A(16x4,f32) * B(4x16,f32)
  return __builtin_amdgcn_wmma_f32_16x16x4_f32(false, a, false, b, (short)0, c,
                                               false, false);
}

__device__ __forceinline__ void wait_async_lds() {
#if defined(HAVE_ASYNC_LDS)
#if __has_builtin(__builtin_amdgcn_s_wait_asynccnt)
  __builtin_amdgcn_s_wait_asynccnt(0);
#else
  asm volatile("s_wait_asynccnt 0x0" ::: "memory");
#endif
#endif
}

// ---------------------------------------------------------------------------
// zero the message accumulators (float4 granularity)
// ---------------------------------------------------------------------------
__global__ void zero4_kernel(float4* __restrict__ p, int n4) {
  int i = blockIdx.x * blockDim.x + threadIdx.x;
  if (i < n4) p[i] = make_float4(0.f, 0.f, 0.f, 0.f);
}

// ---------------------------------------------------------------------------
// K1: h[b,n,i] = (1/sqrt(C)) * sum_m x[b,m,i] * W[m,n]
// rows enumerated r = b*dim + i ; WMMA tiles of 16 rows x 16 outs, K=32.
// h layout: [node][dim][channel]  (channel innermost for edge gathers)
// ---------------------------------------------------------------------------
__global__ __launch_bounds__(256)
void up_kernel(const float* __restrict__ x, const float* __restrict__ w,
               float* __restrict__ h, int dim, int ntiles) {
  const int wave = threadIdx.x >> 5, lane = threadIdx.x & 31;
  const int lm = lane & 15, hi = lane >> 4;
  const int tile = blockIdx.x * 8 + wave;
  if (tile >= ntiles) return;
  const int mB = tile * 16;
  const int row = mB + lm;
  const int b = row / dim, i = row - b * dim;
  for (int nt = 0; nt < 2; ++nt) {
    const int nB = nt * 16;
    v8f acc = {0.f,0.f,0.f,0.f,0.f,0.f,0.f,0.f};
    for (int ks = 0; ks < 8; ++ks) {
      const int kb = ks * 4 + hi * 2;     // this lane's K pair
      v2f a, bb;
      a.x  = x[(b * CC + kb    ) * dim + i];
      a.y  = x[(b * CC + kb + 1) * dim + i];
      bb.x = w[(kb    ) * CC + nB + lm];
      bb.y = w[(kb + 1) * CC + nB + lm];
      acc = wmma4(a, bb, acc);
    }
    for (int r = 0; r < 8; ++r) {
      const int orow = mB + hi * 8 + r;
      h[orow * CC + nB + lm] = acc[r] * INV_C;
    }
  }
}

// ---------------------------------------------------------------------------
// K2: fused edge kernel. 64 edges / block, 256 threads (8 waves).
//   phase 0: async-DMA w_r1/w_r2 into LDS (overlaps with phase 1)
//   phase 1: per-edge geometry + C222 (in LDS)
//   phase 2: radial MLP 8->64->64->352 with WMMA f32 16x16x4, results in LDS
//   phase 3: tensor products, lane=channel, atomicAdd into message buffers
// LDS: ~155 KB of 320 KB per WGP -> 2 workgroups resident.
// ---------------------------------------------------------------------------
#define EB 64
__global__ __launch_bounds__(256)
void edge_kernel(const float* __restrict__ vectors,
                 const int*   __restrict__ senders,
                 const int*   __restrict__ receivers,
                 const float* __restrict__ w_r1,
                 const float* __restrict__ w_r2,
                 const float* __restrict__ w_r3,
                 const float* __restrict__ h0g,
                 const float* __restrict__ h1g,
                 const float* __restrict__ h2g,
                 float* __restrict__ gm0,
                 float* __restrict__ gm1,
                 float* __restrict__ gm2) {
  __shared__ float sC[126];            // C222 (unnorm) + [125]=1/sqrt(ssq/5)
  __shared__ float sBasis[EB][RD];
  __shared__ float sGeom[EB][58];      // s1:0-2 s2:3-7 T:8-16 V:17-31 Q:32-56 flag:57
  __shared__ float sH1[EB][HID];
  __shared__ float sH2[EB][HID];
  __shared__ float sRW[EB][RWN];
  __shared__ __align__(16) float sW1[RD*HID];    // 2 KB,  w_r1 staged
  __shared__ __align__(16) float sW2[HID*HID];   // 16 KB, w_r2 staged

  const int tid = threadIdx.x;
  const int e0 = blockIdx.x * EB;

  // ---- phase 0: async DMA of MLP weights into LDS (ASYNCcnt-tracked) ----
#if defined(HAVE_ASYNC_LDS)
  if (tid < (RD*HID)/4) {              // 128 x b128
    __builtin_amdgcn_global_load_async_to_lds_b128(
        (gv4i*)(w_r1 + tid*4), (lv4i*)(sW1 + tid*4), 0, 0);
  }
  for (int t = tid; t < (HID*HID)/4; t += 256) {   // 1024 x b128
    __builtin_amdgcn_global_load_async_to_lds_b128(
        (gv4i*)(w_r2 + t*4), (lv4i*)(sW2 + t*4), 0, 0);
  }
#else
  for (int t = tid; t < (RD*HID)/4; t += 256)
    ((float4*)sW1)[t] = ((const float4*)w_r1)[t];
  for (int t = tid; t < (HID*HID)/4; t += 256)
    ((float4*)sW2)[t] = ((const float4*)w_r2)[t];
#endif

  // ---- phase 1a: unnormalized C222 (threads 0..124) ----
  if (tid < 125) {
    const int k = tid / 25, m = (tid / 5) % 5, n = tid % 5;
    float S[3][3];
    for (int i = 0; i < 3; ++i)
      for (int j = 0; j < 3; ++j) {
        float a = 0.f;
        for (int t = 0; t < 3; ++t)
          a += c_B[m][i][t] * c_B[n][t][j] + c_B[n][i][t] * c_B[m][t][j];
        S[i][j] = 0.5f * a;
      }
    const float tr3 = (S[0][0] + S[1][1] + S[2][2]) * (1.f / 3.f);
    S[0][0] -= tr3; S[1][1] -= tr3; S[2][2] -= tr3;
    float cv = 0.f;
    for (int i = 0; i < 3; ++i)
      for (int j = 0; j < 3; ++j) cv += c_B[k][i][j] * S[i][j];
    sC[tid] = cv;
  }
  // ---- phase 1b: per-edge geometry + basis (threads 128..191) ----
  if (tid >= 128 && tid < 128 + EB) {
    const int le = tid - 128;
    const int e = e0 + le;
    const float vx = vectors[e*3+0], vy = vectors[e*3+1], vz = vectors[e*3+2];
    const float r = sqrtf(vx*vx + vy*vy + vz*vz);
    const float flag = (r == 0.f) ? 0.f : 1.f;
    const float inv = (r == 0.f) ? 0.f : 1.f / r;
    const float ux = vx*inv, uy = vy*inv, uz = vz*inv;
    // polynomial envelope (p=5) * Bessel basis
    const float u = r * (1.f / CUT);
    const float u5 = u*u*u*u*u;
    float env = 1.f - 21.f*u5 + 35.f*u5*u - 15.f*u5*u*u;
    env = (r < CUT) ? env : 0.f;
    const float sq2rm = 0.63245553f;           // sqrt(2/5)
    for (int nb = 1; nb <= RD; ++nb) {
      const float arg = 3.14159265f * (float)nb * r * (1.f / CUT);
      const float bess = (r > 1e-12f)
          ? sq2rm * __sinf(arg) / r
          : sq2rm * 3.14159265f * (float)nb * (1.f / CUT);
      sBasis[le][nb-1] = bess * env;
    }
    float* G = sGeom[le];
    const float s1x = 1.73205081f*ux, s1y = 1.73205081f*uy, s1z = 1.73205081f*uz;
    G[0]=s1x; G[1]=s1y; G[2]=s1z;
    G[3] = 3.87298335f*ux*uy;
    G[4] = 3.87298335f*uy*uz;
    G[5] = 1.11803399f*(3.f*uz*uz - 1.f);
    G[6] = 3.87298335f*ux*uz;
    G[7] = 1.93649167f*(ux*ux - uy*uy);
    // T[i][j] = sum_m s2_m * B[m][i][j]
    for (int i = 0; i < 3; ++i)
      for (int j = 0; j < 3; ++j) {
        float a = 0.f;
        for (int m = 0; m < 5; ++m) a += G[3+m] * c_B[m][i][j];
        G[8 + i*3 + j] = a;
      }
    // V[m][j] = (B[m] . s1)_j   (B symmetric -> serves paths 6/8)
    for (int m = 0; m < 5; ++m)
      for (int j = 0; j < 3; ++j)
        G[17 + m*3 + j] = c_B[m][0][j]*s1x + c_B[m][1][j]*s1y + c_B[m][2][j]*s1z;
    G[57] = flag;
  }
  wait_async_lds();      // each wave drains its own ASYNCcnt before barrier
  __syncthreads();
  if (tid == 0) {
    float ss = 0.f;
    for (int i = 0; i < 125; ++i) ss += sC[i]*sC[i];
    sC[125] = rsqrtf(ss * 0.2f);
  }
  __syncthreads();
  // ---- Q[k][m] = norm * sum_n C[k,m,n] s2_n (threads 0..63, one edge each)
  if (tid < EB) {
    float* G = sGeom[tid];
    const float f = sC[125];
    for (int k = 0; k < 5; ++k)
      for (int m = 0; m < 5; ++m) {
        float a = 0.f;
        for (int n = 0; n < 5; ++n) a += sC[(k*5+m)*5 + n] * G[3+n];
        G[32 + k*5 + m] = a * f;
      }
  }
  __syncthreads();

  const int wave = tid >> 5, lane = tid & 31;
  const int lm = lane & 15, hi = lane >> 4;

  // ---- phase 2: radial MLP with WMMA f32 16x16x4 ----
  // L1: basis(64x8) @ w_r1(8x64)  -> 16 tiles, 2 per wave (B from LDS)
  for (int t = wave*2; t < wave*2 + 2; ++t) {
    const int mB = (t >> 2) * 16, nB = (t & 3) * 16;
    v8f acc = {0.f,0.f,0.f,0.f,0.f,0.f,0.f,0.f};
    for (int ks = 0; ks < 2; ++ks) {
      const int kb = ks*4 + hi*2;
      v2f a, bb;
      a.x = sBasis[mB+lm][kb];     a.y = sBasis[mB+lm][kb+1];
      bb.x = sW1[kb*HID + nB+lm];  bb.y = sW1[(kb+1)*HID + nB+lm];
      acc = wmma4(a, bb, acc);
    }
    for (int r = 0; r < 8; ++r)
      sH1[mB + hi*8 + r][nB+lm] = nsilu(acc[r] * 0.35355339f);  // /sqrt(8)
  }
  __syncthreads();
  // L2: h1(64x64) @ w_r2(64x64)  (B from LDS)
  for (int t = wave*2; t < wave*2 + 2; ++t) {
    const int mB = (t >> 2) * 16, nB = (t & 3) * 16;
    v8f acc = {0.f,0.f,0.f,0.f,0.f,0.f,0.f,0.f};
    for (int ks = 0; ks < 16; ++ks) {
      const int kb = ks*4 + hi*2;
      v2f a, bb;
      a.x = sH1[mB+lm][kb];        a.y = sH1[mB+lm][kb+1];
      bb.x = sW2[kb*HID + nB+lm];  bb.y = sW2[(kb+1)*HID + nB+lm];
      acc = wmma4(a, bb, acc);
    }
    for (int r = 0; r < 8; ++r)
      sH2[mB + hi*8 + r][nB+lm] = nsilu(acc[r] * 0.125f);       // /sqrt(64)
  }
  __syncthreads();
  // L3: h2(64x64) @ w_r3(64x352) -> 4x22=88 tiles, 11 per wave (B streamed)
  for (int j = 0; j < 11; ++j) {
    const int t = wave*11 + j;
    const int mB = (t / 22) * 16, nB = (t % 22) * 16;
    v8f acc = {0.f,0.f,0.f,0.f,0.f,0.f,0.f,0.f};
    for (int ks = 0; ks < 16; ++ks) {
      const int kb = ks*4 + hi*2;
      v2f a, bb;
      a.x = sH2[mB+lm][kb];        a.y = sH2[mB+lm][kb+1];
      bb.x = w_r3[kb*RWN + nB+lm]; bb.y = w_r3[(kb+1)*RWN + nB+lm];
      acc = wmma4(a, bb, acc);
    }
    for (int r = 0; r < 8; ++r) {
      const int row = mB + hi*8 + r;
      sRW[row][nB+lm] = acc[r] * 0.125f * EPS_W * sGeom[row][57];
    }
  }
  __syncthreads();

  // ---- phase 3: tensor products + scatter (lane = channel) ----
  const int c = lane;
  for (int it = 0; it < EB/8; ++it) {
    const int le = it*8 + wave;
    const int e = e0 + le;
    const int snd = senders[e], rcv = receivers[e];
    if (it + 1 < EB/8) {           // global_prefetch next sender's features
      const int sn = senders[e0 + (it+1)*8 + wave];
      __builtin_prefetch(&h1g[(sn*3)*CC + c], 0, 0);
      __builtin_prefetch(&h2g[(sn*5)*CC + c], 0, 0);
    }
    const float a0 = h0g[snd*CC + c];
    float a1[3], a2[5];
    for (int i = 0; i < 3; ++i) a1[i] = h1g[(snd*3 + i)*CC + c];
    for (int k = 0; k < 5; ++k) a2[k] = h2g[(snd*5 + k)*CC + c];
    const float* G = sGeom[le];
    const float* W = sRW[le];
    float w[NP];
    for (int p = 0; p < NP; ++p) w[p] = W[p*CC + c];

    // ---- m0 ----
    float* p0 = gm0 + (rcv*3)*CC + c;
    atomicAdd(p0 + 0*CC, a0 * w[0]);
    const float d1 = a1[0]*G[0] + a1[1]*G[1] + a1[2]*G[2];
    atomicAdd(p0 + 1*CC, d1 * w[1] * 0.57735027f);              // /sqrt(3)
    float d2 = 0.f;
    for (int k = 0; k < 5; ++k) d2 += a2[k]*G[3+k];
    atomicAdd(p0 + 2*CC, d2 * w[2] * 0.44721360f);              // /sqrt(5)

    // ---- m1 ----
    const float c12 = 0.77459667f;                              // sqrt(3/5)
    float* p1 = gm1 + (rcv*4*3)*CC + c;
    for (int j = 0; j < 3; ++j) {
      atomicAdd(p1 + (0*3+j)*CC, a0 * G[j] * w[3]);             // path3
      atomicAdd(p1 + (1*3+j)*CC, a1[j] * w[4]);                 // path4
      float t5 = a1[0]*G[8+0*3+j] + a1[1]*G[8+1*3+j] + a1[2]*G[8+2*3+j];
      atomicAdd(p1 + (2*3+j)*CC, t5 * (c12 * w[5]));            // path5
      float t6 = 0.f;
      for (int m = 0; m < 5; ++m) t6 += a2[m]*G[17+m*3+j];
      atomicAdd(p1 + (3*3+j)*CC, t6 * (c12 * w[6]));            // path6
    }

    // ---- m2 ----
    float* p2 = gm2 + (rcv*4*5)*CC + c;
    for (int k = 0; k < 5; ++k) {
      atomicAdd(p2 + (0*5+k)*CC, a0 * G[3+k] * w[7]);           // path7
      float t8 = a1[0]*G[17+k*3+0] + a1[1]*G[17+k*3+1] + a1[2]*G[17+k*3+2];
      atomicAdd(p2 + (1*5+k)*CC, t8 * w[8]);                    // path8
      atomicAdd(p2 + (2*5+k)*CC, a2[k] * w[9]);                 // path9
      float t10 = 0.f;
      for (int m = 0; m < 5; ++m) t10 += a2[m]*G[32+k*5+m];
      atomicAdd(p2 + (3*5+k)*CC, t10 * w[10]);                  // path10
    }
  }
}

// ---------------------------------------------------------------------------
// K3: per-node output. One wave per node, lane = output channel o.
// Message reads are lane-uniform (broadcast); weight reads coalesced.
// ---------------------------------------------------------------------------
__global__ __launch_bounds__(256)
void node_out_kernel(const float* __restrict__ x0, const float* __restrict__ x1,
                     const float* __restrict__ x2,
                     const float* __restrict__ gm0, const float* __restrict__ gm1,
                     const float* __restrict__ gm2,
                     const float* __restrict__ w_d0s, const float* __restrict__ w_d0g,
                     const float* __restrict__ w_d1,  const float* __restrict__ w_d2,
                     const float* __restrict__ w_s0s, const float* __restrict__ w_s0g,
                     const float* __restrict__ w_s1,  const float* __restrict__ w_s2,
                     float* __restrict__ out) {
  const int wave = threadIdx.x >> 5, o = threadIdx.x & 31;
  const int b = blockIdx.x * 8 + wave;
  if (b >= NN) return;

  // d0: m0f(96) @ w_d0s / w_d0g
  float d0s = 0.f, d0g1 = 0.f, d0g2 = 0.f;
  const float* m0 = gm0 + b*3*CC;
  for (int k = 0; k < 3*CC; ++k) {
    const float v = m0[k];
    d0s  += v * w_d0s[k*CC + o];
    d0g1 += v * w_d0g[k*2*CC + o];
    d0g2 += v * w_d0g[k*2*CC + CC + o];
  }
  // d1 / d2
  float d1v[3] = {0.f,0.f,0.f};
  const float* m1 = gm1 + b*4*3*CC;
  for (int k = 0; k < 4*CC; ++k) {
    const int slot = k >> 5, cc = k & 31;
    const float wk = w_d1[k*CC + o];
    for (int i = 0; i < 3; ++i) d1v[i] += m1[(slot*3+i)*CC + cc] * wk;
  }
  float d2v[5] = {0.f,0.f,0.f,0.f,0.f};
  const float* m2 = gm2 + b*4*5*CC;
  for (int k = 0; k < 4*CC; ++k) {
    const int slot = k >> 5, cc = k & 31;
    const float wk = w_d2[k*CC + o];
    for (int i = 0; i < 5; ++i) d2v[i] += m2[(slot*5+i)*CC + cc] * wk;
  }
  // species self-interaction (counts fixed: NN/4 each)
  const int sp = b / (NN/4);
  float ss = 0.f, sg1 = 0.f, sg2 = 0.f;
  float k1v[3] = {0.f,0.f,0.f};
  float k2v[5] = {0.f,0.f,0.f,0.f,0.f};
  for (int i = 0; i < CC; ++i) {
    const float xv = x0[b*CC + i];
    ss  += xv * w_s0s[sp*CC*CC + i*CC + o];
    sg1 += xv * w_s0g[sp*CC*2*CC + i*2*CC + o];
    sg2 += xv * w_s0g[sp*CC*2*CC + i*2*CC + CC + o];
    const float w1i = w_s1[sp*CC*CC + i*CC + o];
    const float w2i = w_s2[sp*CC*CC + i*CC + o];
    for (int d = 0; d < 3; ++d) k1v[d] += x1[(b*CC+i)*3 + d] * w1i;
    for (int d = 0; d < 5; ++d) k2v[d] += x2[(b*CC+i)*5 + d] * w2i;
  }
  const float i96  = 0.10206207f;   // 1/sqrt(3C)
  const float i128 = 0.08838835f;   // 1/sqrt(4C)
  const float scal = d0s*i96 + ss*INV_C;
  const float g1   = nsilu(d0g1*i96 + sg1*INV_C);
  const float g2   = nsilu(d0g2*i96 + sg2*INV_C);
  out[b*CC + o] = nsilu(scal);
  float* o1 = out + NN*CC + (b*CC + o)*3;
  for (int i = 0; i < 3; ++i) o1[i] = (d1v[i]*i128 + k1v[i]*INV_C) * g1;
  float* o2 = out + NN*CC*4 + (b*CC + o)*5;
  for (int i = 0; i < 5; ++i) o2[i] = (d2v[i]*i128 + k2v[i]*INV_C) * g2;
}

// ---------------------------------------------------------------------------
extern "C" void kernel_launch(void* const* d_in, const int* in_sizes, int n_in,
                              void* d_out, int out_size, void* d_ws, size_t ws_size,
                              hipStream_t stream) {
  (void)in_sizes; (void)n_in; (void)out_size; (void)ws_size;
  const float* vectors  = (const float*)d_in[0];
  const float* x0       = (const float*)d_in[1];
  const float* x1       = (const float*)d_in[2];
  const float* x2       = (const float*)d_in[3];
  const int*   senders  = (const int*)  d_in[5];
  const int*   receivers= (const int*)  d_in[6];
  const float* w_up0    = (const float*)d_in[8];
  const float* w_up1    = (const float*)d_in[9];
  const float* w_up2    = (const float*)d_in[10];
  const float* w_r1     = (const float*)d_in[11];
  const float* w_r2     = (const float*)d_in[12];
  const float* w_r3     = (const float*)d_in[13];
  const float* w_d0s    = (const float*)d_in[14];
  const float* w_d0g    = (const float*)d_in[15];
  const float* w_d1     = (const float*)d_in[16];
  const float* w_d2     = (const float*)d_in[17];
  const float* w_s0s    = (const float*)d_in[18];
  const float* w_s0g    = (const float*)d_in[19];
  const float* w_s1     = (const float*)d_in[20];
  const float* w_s2     = (const float*)d_in[21];

  // workspace layout (floats): h buffers then message accumulators (~113 MB)
  float* ws  = (float*)d_ws;
  float* h0g = ws;                       // NN*32
  float* h1g = h0g + NN*CC;              // NN*96
  float* h2g = h1g + NN*3*CC;            // NN*160
  float* gm0 = h2g + NN*5*CC;            // NN*96
  float* gm1 = gm0 + NN*3*CC;            // NN*384
  float* gm2 = gm1 + NN*4*3*CC;          // NN*640

  // zero message accumulators every call (atomics accumulate; deterministic)
  const int nz4 = (NN*(3 + 4*3 + 4*5)*CC) / 4;   // 5,600,000 float4
  zero4_kernel<<<(nz4 + 255)/256, 256, 0, stream>>>((float4*)gm0, nz4);

  up_kernel<<<(NN/16 + 7)/8,     256, 0, stream>>>(x0, w_up0, h0g, 1, NN/16);
  up_kernel<<<(NN*3/16 + 7)/8,   256, 0, stream>>>(x1, w_up1, h1g, 3, NN*3/16);
  up_kernel<<<(NN*5/16 + 7)/8,   256, 0, stream>>>(x2, w_up2, h2g, 5, NN*5/16);

  edge_kernel<<<NE/EB, 256, 0, stream>>>(vectors, senders, receivers,
                                         w_r1, w_r2, w_r3,
                                         h0g, h1g, h2g, gm0, gm1, gm2);

  node_out_kernel<<<(NN + 7)/8, 256, 0, stream>>>(x0, x1, x2, gm0, gm1, gm2,
                                                  w_d0s, w_d0g, w_d1, w_d2,
                                                  w_s0s, w_s0g, w_s1, w_s2,
                                                  (float*)d_out);
}